// AnchorTargetLayer_25477745999924
// MI455X (gfx1250) — compile-verified
//
#include <hip/hip_runtime.h>
#include <hip/hip_bf16.h>
#include <stdint.h>

#define BATCH   8
#define NG      128          // gt boxes per batch
#define IH      64
#define IW      64
#define NA      9            // base anchors
#define NSHIFT  (IH*IW)
#define NK      (NSHIFT*NA)  // 36864 anchors total
#define ROWS    128          // k-rows per block
#define THREADS 256          // 8 waves (wave32)

// overlaps[b][k][g] : IoU of shifted anchor k vs gt box g of batch b.
__global__ __launch_bounds__(THREADS)
void anchor_overlaps_kernel(const float* __restrict__ gt_boxes,  // [B,128,5]
                            const float* __restrict__ im_info,   // [B,3]
                            const float* __restrict__ anchors,   // [9,4]
                            float* __restrict__ out)             // [B,NK,128]
{
    // First shared object -> LDS offset 0 (used by async-to-LDS staging).
    __shared__ float gtbuf[NG * 5];                 // 2560 B slab for this batch
    __shared__ float ax1s[NA], ay1s[NA], ax2s[NA], ay2s[NA], areas[NA];
    __shared__ int   azero[NA];

    const int tid  = threadIdx.x;
    const int lane = tid & 31;
    const int wave = tid >> 5;

    const int blocksPerB = NK / ROWS;               // 288
    const int b  = blockIdx.x / blocksPerB;
    const int k0 = (blockIdx.x % blocksPerB) * ROWS;

    // ---- CDNA5 async copy: gt_boxes[b] (640 dwords) -> LDS, 160 lanes x 16B ----
    if (tid < (NG * 5 * 4) / 16) {
        unsigned ldsoff = (unsigned)(uint64_t)(uintptr_t)gtbuf + (unsigned)(tid * 16);
        unsigned goff   = (unsigned)(b * (NG * 5 * 4)) + (unsigned)(tid * 16);
        asm volatile("global_load_async_to_lds_b128 %0, %1, %2"
                     :: "v"(ldsoff), "v"(goff), "s"(gt_boxes)
                     : "memory");
    }
    // Shift-invariant anchor geometry (width/height/area/zero are shift-free).
    if (tid < NA) {
        float x1 = anchors[tid * 4 + 0], y1 = anchors[tid * 4 + 1];
        float x2 = anchors[tid * 4 + 2], y2 = anchors[tid * 4 + 3];
        float w = x2 - x1 + 1.f, h = y2 - y1 + 1.f;
        ax1s[tid] = x1; ay1s[tid] = y1; ax2s[tid] = x2; ay2s[tid] = y2;
        areas[tid] = w * h;
        azero[tid] = (w == 1.f) && (h == 1.f);
    }
    asm volatile("s_wait_asynccnt 0" ::: "memory");
    __syncthreads();

    const float im_h = im_info[0];   // im_info[0,0]
    const float im_w = im_info[1];   // im_info[0,1]

    // Each lane caches its 4 gt boxes (g = 4*lane .. 4*lane+3) in registers.
    float gx1[4], gy1[4], gx2[4], gy2[4], garea[4];
    bool  gzero[4];
#pragma unroll
    for (int j = 0; j < 4; ++j) {
        const float* p = &gtbuf[(lane * 4 + j) * 5];
        gx1[j] = p[0]; gy1[j] = p[1]; gx2[j] = p[2]; gy2[j] = p[3];
        float gw = gx2[j] - gx1[j] + 1.f;
        float gh = gy2[j] - gy1[j] + 1.f;
        garea[j] = gw * gh;
        gzero[j] = (gw == 1.f) && (gh == 1.f);
    }

    float4* __restrict__ out4 = reinterpret_cast<float4*>(out);

    for (int r = wave; r < ROWS; r += THREADS / 32) {
        // k is identical across the wave; pin it to an SGPR for scalar branching.
        int k = __builtin_amdgcn_readfirstlane(k0 + r);
        int s = k / 9;                 // shift index (row-major over HxW)
        int a = k - s * 9;             // base-anchor index
        int x = s & (IW - 1);
        int y = s >> 6;
        float sx = (float)(x * 16);    // FEAT_STRIDE = 16
        float sy = (float)(y * 16);

        float ax1 = ax1s[a] + sx, ay1 = ay1s[a] + sy;
        float ax2 = ax2s[a] + sx, ay2 = ay2s[a] + sy;
        float aarea = areas[a];

        bool inside = (ax1 >= 0.f) & (ay1 >= 0.f) & (ax2 < im_w) & (ay2 < im_h);
        const int idx = (b * NK + k) * (NG / 4) + lane;   // float4 index

        float4 o;
        if (azero[a] || !inside) {
            // a_zero / !inside override everything with -1 (uniform fast path).
            o = make_float4(-1.f, -1.f, -1.f, -1.f);
        } else {
            float ov[4];
#pragma unroll
            for (int j = 0; j < 4; ++j) {
                float ix1 = fmaxf(ax1, gx1[j]);
                float iy1 = fmaxf(ay1, gy1[j]);
                float ix2 = fminf(ax2, gx2[j]);
                float iy2 = fminf(ay2, gy2[j]);
                float iw  = fmaxf(ix2 - ix1 + 1.f, 0.f);
                float ih  = fmaxf(iy2 - iy1 + 1.f, 0.f);
                float inter = iw * ih;
                float ua = aarea + garea[j] - inter;
                float v  = inter * __builtin_amdgcn_rcpf(ua);  // v_rcp_f32, IoU-grade accuracy
                ov[j] = gzero[j] ? 0.f : v;
            }
            o = make_float4(ov[0], ov[1], ov[2], ov[3]);
        }
        out4[idx] = o;   // global_store_b128, 512B contiguous per wave-row
    }
}

extern "C" void kernel_launch(void* const* d_in, const int* in_sizes, int n_in,
                              void* d_out, int out_size, void* d_ws, size_t ws_size,
                              hipStream_t stream) {
    // Input order: rpn_cls_score, gt_boxes, im_info, num_boxes, anchors.
    // rpn_cls_score (only provides H,W=64,64) and num_boxes are unused.
    const float* gt_boxes = (const float*)d_in[1];
    const float* im_info  = (const float*)d_in[2];
    const float* anchors  = (const float*)d_in[4];
    float* out = (float*)d_out;

    dim3 grid(BATCH * (NK / ROWS));   // 8 * 288 = 2304 blocks
    anchor_overlaps_kernel<<<grid, THREADS, 0, stream>>>(gt_boxes, im_info, anchors, out);
}